// GAT_24300924961042
// MI455X (gfx1250) — compile-verified
//
#include <hip/hip_runtime.h>
#include <hip/hip_bf16.h>
#include <math.h>

typedef __attribute__((ext_vector_type(2))) float v2f;
typedef __attribute__((ext_vector_type(8))) float v8f;

#define N_NODES   50000
#define N_EDGES   1600000
#define NE_TOT    (N_EDGES + N_NODES)
#define D_IN      128
#define H1        64
#define H2        128
#define NGRAPHS   64
#define NEG_SLOPE 0.2f

// ---------- float <-> monotonic uint encoding (for atomic float max) ----------
__device__ __forceinline__ unsigned fenc(float f) {
    unsigned u = __float_as_uint(f);
    return (u & 0x80000000u) ? ~u : (u | 0x80000000u);
}
__device__ __forceinline__ float fdec(unsigned u) {
    u = (u & 0x80000000u) ? (~u) : (u & 0x7FFFFFFFu);
    return __uint_as_float(u);
}
#define ENC_NEG_INF 0x007FFFFFu   // fenc(-INFINITY)

// ---------- WMMA GEMM: C[N,M] = A[N,K] @ B[K,M] (+bias)(+relu) ----------
// One wave per 16x16 output tile, K stepped by 4 via v_wmma_f32_16x16x4_f32.
// Requires Nrows % 16 == 0, M % 16 == 0, K % 4 == 0 (true for all uses here).
__global__ void gemm_wmma_f32(const float* __restrict__ A, const float* __restrict__ B,
                              float* __restrict__ C, int Nrows, int K, int M,
                              const float* __restrict__ bias, int relu)
{
    int waveId = (int)((blockIdx.x * (long)blockDim.x + threadIdx.x) >> 5);
    int lane   = threadIdx.x & 31;
    int tilesM = M >> 4;
    int tilesN = Nrows >> 4;
    if (waveId >= tilesM * tilesN) return;          // whole wave exits together (EXEC stays all-1)
    int tm = waveId / tilesM;
    int tn = waveId % tilesM;

    int half = lane >> 4;                           // 0: K+0/K+1, 1: K+2/K+3
    int l15  = lane & 15;
    int arow = tm * 16 + l15;                       // A: M striped across lanes
    int bcol = tn * 16 + l15;                       // B: N striped across lanes

    v8f c = {0.f, 0.f, 0.f, 0.f, 0.f, 0.f, 0.f, 0.f};
    for (int k0 = 0; k0 < K; k0 += 4) {
        int kb = k0 + half * 2;
        v2f a, b;
        a.x = A[(long)arow * K + kb];
        a.y = A[(long)arow * K + kb + 1];
        b.x = B[(long)kb * M + bcol];
        b.y = B[(long)(kb + 1) * M + bcol];
        c = __builtin_amdgcn_wmma_f32_16x16x4_f32(false, a, false, b, (short)0, c, false, false);
    }
#pragma unroll
    for (int i = 0; i < 8; ++i) {                   // C/D: VGPR i -> row i (lanes 0-15), row i+8 (lanes 16-31)
        int row = tm * 16 + i + half * 8;
        float v = c[i];
        if (bias) v += bias[bcol];
        if (relu) v = fmaxf(v, 0.f);
        C[(long)row * M + bcol] = v;
    }
}

// ---------- per-node alpha dots + per-layer state init (wave per node) ----------
__global__ void node_alpha_init(const float* __restrict__ h, int F,
                                const float* __restrict__ a_src, const float* __restrict__ a_dst,
                                float* __restrict__ alpha_s, float* __restrict__ alpha_d,
                                unsigned* __restrict__ segmax_enc, float* __restrict__ denom,
                                float* __restrict__ out_accum)
{
    int wave = (int)((blockIdx.x * (long)blockDim.x + threadIdx.x) >> 5);
    int lane = threadIdx.x & 31;
    if (wave >= N_NODES) return;
    const float* row = h + (long)wave * F;
    float ss = 0.f, sd = 0.f;
    for (int f = lane; f < F; f += 32) {
        float v = row[f];
        ss += v * a_src[f];
        sd += v * a_dst[f];
        out_accum[(long)wave * F + f] = 0.f;        // zero the scatter accumulator
    }
    for (int off = 16; off > 0; off >>= 1) {
        ss += __shfl_down(ss, off, 32);
        sd += __shfl_down(sd, off, 32);
    }
    if (lane == 0) {
        alpha_s[wave] = ss;
        alpha_d[wave] = sd;
        segmax_enc[wave] = ENC_NEG_INF;
        denom[wave] = 0.f;
    }
}

// ---------- edge pass 1: e = LeakyReLU(as[src]+ad[dst]); segment max ----------
__global__ void edge_max(const int* __restrict__ esrc, const int* __restrict__ edst,
                         const float* __restrict__ alpha_s, const float* __restrict__ alpha_d,
                         float* __restrict__ ebuf, unsigned* __restrict__ segmax_enc)
{
    long e = blockIdx.x * (long)blockDim.x + threadIdx.x;
    if (e >= NE_TOT) return;
    int s, d;
    if (e < N_EDGES) { s = esrc[e]; d = edst[e]; }
    else             { s = d = (int)(e - N_EDGES); }   // implicit self loops
    float v = alpha_s[s] + alpha_d[d];
    v = (v > 0.f) ? v : NEG_SLOPE * v;
    ebuf[e] = v;
    atomicMax(&segmax_enc[d], fenc(v));
}

// ---------- edge pass 2: ex = exp(e - max[dst]); segment sum ----------
__global__ void edge_expsum(const int* __restrict__ edst,
                            float* __restrict__ ebuf,
                            const unsigned* __restrict__ segmax_enc,
                            float* __restrict__ denom)
{
    long e = blockIdx.x * (long)blockDim.x + threadIdx.x;
    if (e >= NE_TOT) return;
    int d = (e < N_EDGES) ? edst[e] : (int)(e - N_EDGES);
    float m  = fdec(segmax_enc[d]);
    float ex = __expf(ebuf[e] - m);
    ebuf[e] = ex;
    atomicAdd(&denom[d], ex);
}

// ---------- edge pass 3: out[dst] += h[src] * (ex / denom[dst])  (wave per edge) ----------
__global__ void edge_scatter(const int* __restrict__ esrc, const int* __restrict__ edst, int F,
                             const float* __restrict__ ebuf, const float* __restrict__ denom,
                             const float* __restrict__ h, float* __restrict__ out)
{
    long wave = (blockIdx.x * (long)blockDim.x + threadIdx.x) >> 5;
    int lane = threadIdx.x & 31;
    if (wave >= NE_TOT) return;
    int s, d;
    if (wave < N_EDGES) { s = esrc[wave]; d = edst[wave]; }
    else                { s = d = (int)(wave - N_EDGES); }
    float a = ebuf[wave] / (denom[d] + 1e-16f);
    const float* hr = h + (long)s * F;
    float* orow = out + (long)d * F;
    for (int f = lane; f < F; f += 32)
        atomicAdd(&orow[f], hr[f] * a);
}

// ---------- epilogue: x = relu(x + bias)  (in place) ----------
__global__ void bias_relu(float* __restrict__ x, const float* __restrict__ bias,
                          int Fmask, long total)
{
    long i = blockIdx.x * (long)blockDim.x + threadIdx.x;
    if (i >= total) return;
    int col = (int)(i & (long)Fmask);
    x[i] = fmaxf(x[i] + bias[col], 0.f);
}

// ---------- pooling ----------
__global__ void pool_zero(float* __restrict__ pooled, float* __restrict__ cnts)
{
    int i = blockIdx.x * blockDim.x + threadIdx.x;
    if (i < NGRAPHS * H2) pooled[i] = 0.f;
    if (i < NGRAPHS) cnts[i] = 0.f;
}

__global__ void pool_accum(const int* __restrict__ batch, const float* __restrict__ h2,
                           float* __restrict__ pooled, float* __restrict__ cnts)
{
    int wave = (int)((blockIdx.x * (long)blockDim.x + threadIdx.x) >> 5);
    int lane = threadIdx.x & 31;
    if (wave >= N_NODES) return;
    int g = batch[wave];
    const float* hr = h2 + (long)wave * H2;
    for (int f = lane; f < H2; f += 32)
        atomicAdd(&pooled[(long)g * H2 + f], hr[f]);
    if (lane == 0) atomicAdd(&cnts[g], 1.f);
}

__global__ void pool_div(float* __restrict__ pooled, const float* __restrict__ cnts)
{
    int i = blockIdx.x * blockDim.x + threadIdx.x;
    if (i >= NGRAPHS * H2) return;
    pooled[i] /= fmaxf(cnts[i / H2], 1.f);
}

// ======================================================================
extern "C" void kernel_launch(void* const* d_in, const int* in_sizes, int n_in,
                              void* d_out, int out_size, void* d_ws, size_t ws_size,
                              hipStream_t stream) {
    (void)in_sizes; (void)n_in; (void)out_size; (void)ws_size;

    const float* x       = (const float*)d_in[0];
    const int*   eidx    = (const int*)  d_in[1];   // [2,E]: row0=src, row1=dst
    const int*   batch   = (const int*)  d_in[2];
    const float* W1      = (const float*)d_in[3];
    const float* a_src1  = (const float*)d_in[4];
    const float* a_dst1  = (const float*)d_in[5];
    const float* b1      = (const float*)d_in[6];
    const float* W2      = (const float*)d_in[7];
    const float* a_src2  = (const float*)d_in[8];
    const float* a_dst2  = (const float*)d_in[9];
    const float* b2      = (const float*)d_in[10];
    const float* Wfc     = (const float*)d_in[11];
    const float* bfc     = (const float*)d_in[12];
    float* out = (float*)d_out;

    const int* esrc = eidx;
    const int* edst = eidx + N_EDGES;

    // ---- workspace layout ----
    char* ws = (char*)d_ws;
    size_t off = 0;
    auto carve = [&](size_t bytes) -> char* {
        char* p = ws + off;
        off = (off + bytes + 255) & ~(size_t)255;
        return p;
    };
    float*    h1      = (float*)   carve((size_t)N_NODES * H1 * sizeof(float));
    float*    o1      = (float*)   carve((size_t)N_NODES * H1 * sizeof(float));
    float*    h2lin   = (float*)   carve((size_t)N_NODES * H2 * sizeof(float));
    float*    o2      = (float*)   carve((size_t)N_NODES * H2 * sizeof(float));
    float*    als     = (float*)   carve((size_t)N_NODES * sizeof(float));
    float*    ald     = (float*)   carve((size_t)N_NODES * sizeof(float));
    unsigned* segmax  = (unsigned*)carve((size_t)N_NODES * sizeof(unsigned));
    float*    denom   = (float*)   carve((size_t)N_NODES * sizeof(float));
    float*    ebuf    = (float*)   carve((size_t)NE_TOT * sizeof(float));
    float*    pooled  = (float*)   carve((size_t)NGRAPHS * H2 * sizeof(float));
    float*    cnts    = (float*)   carve((size_t)NGRAPHS * sizeof(float));

    const int BLK = 256;
    auto cdiv = [](long a, long b) { return (int)((a + b - 1) / b); };

    // ================= Layer 1 (F = 64) =================
    {   // h1 = x @ W1        (50000x128 @ 128x64)
        long waves = (N_NODES / 16) * (H1 / 16);
        gemm_wmma_f32<<<cdiv(waves * 32, BLK), BLK, 0, stream>>>(x, W1, h1, N_NODES, D_IN, H1, nullptr, 0);
    }
    node_alpha_init<<<cdiv((long)N_NODES * 32, BLK), BLK, 0, stream>>>(
        h1, H1, a_src1, a_dst1, als, ald, segmax, denom, o1);
    edge_max   <<<cdiv(NE_TOT, BLK), BLK, 0, stream>>>(esrc, edst, als, ald, ebuf, segmax);
    edge_expsum<<<cdiv(NE_TOT, BLK), BLK, 0, stream>>>(edst, ebuf, segmax, denom);
    edge_scatter<<<cdiv((long)NE_TOT * 32, BLK), BLK, 0, stream>>>(
        esrc, edst, H1, ebuf, denom, h1, o1);
    bias_relu<<<cdiv((long)N_NODES * H1, BLK), BLK, 0, stream>>>(o1, b1, H1 - 1, (long)N_NODES * H1);

    // ================= Layer 2 (F = 128) =================
    {   // h2lin = o1 @ W2    (50000x64 @ 64x128)
        long waves = (N_NODES / 16) * (H2 / 16);
        gemm_wmma_f32<<<cdiv(waves * 32, BLK), BLK, 0, stream>>>(o1, W2, h2lin, N_NODES, H1, H2, nullptr, 0);
    }
    node_alpha_init<<<cdiv((long)N_NODES * 32, BLK), BLK, 0, stream>>>(
        h2lin, H2, a_src2, a_dst2, als, ald, segmax, denom, o2);
    edge_max   <<<cdiv(NE_TOT, BLK), BLK, 0, stream>>>(esrc, edst, als, ald, ebuf, segmax);
    edge_expsum<<<cdiv(NE_TOT, BLK), BLK, 0, stream>>>(edst, ebuf, segmax, denom);
    edge_scatter<<<cdiv((long)NE_TOT * 32, BLK), BLK, 0, stream>>>(
        esrc, edst, H2, ebuf, denom, h2lin, o2);
    bias_relu<<<cdiv((long)N_NODES * H2, BLK), BLK, 0, stream>>>(o2, b2, H2 - 1, (long)N_NODES * H2);

    // ================= Global mean pool =================
    pool_zero <<<cdiv(NGRAPHS * H2, BLK), BLK, 0, stream>>>(pooled, cnts);
    pool_accum<<<cdiv((long)N_NODES * 32, BLK), BLK, 0, stream>>>(batch, o2, pooled, cnts);
    pool_div  <<<cdiv(NGRAPHS * H2, BLK), BLK, 0, stream>>>(pooled, cnts);

    // ================= FC: out = pooled @ Wfc + bfc   (64x128 @ 128x64) =================
    {
        long waves = (NGRAPHS / 16) * (H1 / 16);
        gemm_wmma_f32<<<cdiv(waves * 32, BLK), BLK, 0, stream>>>(pooled, Wfc, out, NGRAPHS, H2, H1, bfc, 0);
    }
}